// CrossGAT_72679436583446
// MI455X (gfx1250) — compile-verified
//
#include <hip/hip_runtime.h>
#include <math.h>

// ---------------------------------------------------------------------------
// GAT (8 heads x 16) + GRU cell, MI455X / gfx1250, wave32.
//  - Dense GEMMs: V_WMMA_F32_16X16X4_F32 (full fp32 precision; GEMMs are a
//    small fraction of runtime, edge scatter dominates, so fp32 is free).
//  - k1 stages the 64KB packed weight matrix into LDS with
//    global_load_async_to_lds_b128 (+ s_wait_asynccnt), B-frags via DS.
//  - k6 (dominant phase): 1 wave/edge, b128 gathers, L2-resident atomics.
// ---------------------------------------------------------------------------

#define ALPHA  0.2f
#define NN     50000
#define NE     800000
#define FDIM   128     // NHID
#define NH     8       // heads
#define DHD    16      // per-head dim
#define G3     384     // 3*NHID

typedef float v2f __attribute__((ext_vector_type(2)));
typedef float v8f __attribute__((ext_vector_type(8)));

// D = A(16x4) * B(4x16) + C, fp32.
__device__ __forceinline__ v8f wmma_f32(v2f a, v2f b, v8f c) {
  return __builtin_amdgcn_wmma_f32_16x16x4_f32(false, a, false, b, (short)0, c,
                                               false, false);
}

__device__ __forceinline__ float lrelu(float x) { return x > 0.f ? x : ALPHA * x; }
__device__ __forceinline__ float sigm(float x)  { return 1.f / (1.f + __expf(-x)); }
__device__ __forceinline__ float tanh_fast(float x) {
  x = fminf(15.f, fmaxf(-15.f, x));
  float e = __expf(2.f * x);
  return (e - 1.f) / (e + 1.f);
}

// Float atomic max via monotone int/uint mapping (no CAS loop).
__device__ __forceinline__ void atomicMaxF(float* addr, float v) {
  if (v >= 0.f) atomicMax((int*)addr, __float_as_int(v));
  else          atomicMin((unsigned int*)addr, __float_as_uint(v));
}

// --------------------------- k0: weight repack ------------------------------
// Wflat[f][c]       = W[c/16][f][c%16]            (128x128, K = f)
// wiP[(k*128+c)*3+g] = w_ih[(g*128+c)*128 + k]     (gate-interleaved, b96 frag)
__global__ __launch_bounds__(256) void k0_pack(
    const float* __restrict__ W, const float* __restrict__ w_ih,
    const float* __restrict__ w_hh, float* __restrict__ Wflat,
    float* __restrict__ wiP, float* __restrict__ whP) {
  int idx = blockIdx.x * 256 + threadIdx.x;
  if (idx < FDIM * FDIM) {
    int f = idx >> 7, c = idx & 127;
    Wflat[idx] = W[(c >> 4) * (FDIM * DHD) + f * DHD + (c & 15)];
  }
  if (idx < FDIM * G3) {              // 49152
    int g   = idx % 3;
    int col = (idx / 3) & 127;
    int k   = idx / G3;
    wiP[idx] = w_ih[(g * FDIM + col) * FDIM + k];
    whP[idx] = w_hh[(g * FDIM + col) * FDIM + k];
  }
}

// --------------------------- k1: Wh = h @ Wflat -----------------------------
// 8 waves/block; block owns 16 node-rows; wave w owns cols [16w,16w+16).
// Wflat staged in LDS via async global->LDS b128 DMA; B-frags from DS.
__global__ __launch_bounds__(256) void k1_wh(
    const float* __restrict__ h, const float* __restrict__ Wflat,
    float* __restrict__ Wh) {
  __shared__ __align__(16) float hA[16][132];   // pad: conflict-free col reads
  __shared__ __align__(16) float Wl[128][132];  // 67.6KB staged Wflat (padded)
  int tid = threadIdx.x;
  int mbase = blockIdx.x * 16;

  // Async-stage Wflat (16384 floats = 4096 x b128) into padded LDS.
  {
    unsigned lbase = (unsigned)(unsigned long long)&Wl[0][0];
    for (int i = tid; i < 4096; i += 256) {
      unsigned row  = (unsigned)i >> 5;
      unsigned col4 = ((unsigned)i & 31u) << 2;
      unsigned loff = lbase + (row * 132u + col4) * 4u;
      unsigned goff = (unsigned)i * 16u;
      asm volatile("global_load_async_to_lds_b128 %0, %1, %2"
                   :: "v"(loff), "v"(goff), "s"(Wflat)
                   : "memory");
    }
  }
  // Stage h tile (2048 floats = 512 x float4).
  for (int i = tid; i < 512; i += 256) {
    int r = i >> 5, c4 = (i & 31) << 2;
    *(float4*)&hA[r][c4] = *(const float4*)&h[(mbase + r) * FDIM + c4];
  }
  asm volatile("s_wait_asynccnt 0x0" ::: "memory");
  __syncthreads();

  int wave = tid >> 5, lane = tid & 31;
  int half = lane >> 4, l16 = lane & 15;
  int c0 = wave * 16;
  v8f acc = {};
  for (int k0 = 0; k0 < FDIM; k0 += 4) {
    int ks = k0 + 2 * half;                 // ISA A-layout: K0/K1 | K2/K3
    v2f a, b;
    a.x = hA[l16][ks];       a.y = hA[l16][ks + 1];
    b.x = Wl[ks][c0 + l16];  b.y = Wl[ks + 1][c0 + l16];
    acc = wmma_f32(a, b, acc);
  }
#pragma unroll
  for (int r = 0; r < 8; ++r) {             // D: VGPR r -> row r+8*half
    int M = r + 8 * half;
    Wh[(mbase + M) * FDIM + c0 + l16] = acc[r];
  }
}

// --------------------------- k2: attention scores ---------------------------
__global__ __launch_bounds__(256) void k2_scores(
    const float* __restrict__ Wh, const float* __restrict__ a,
    float* __restrict__ s_src, float* __restrict__ s_dst) {
  int idx = blockIdx.x * 256 + threadIdx.x;
  if (idx >= NN * NH) return;
  int n = idx >> 3, hh = idx & 7;
  const float* w  = Wh + n * FDIM + hh * DHD;
  const float* a1 = a + hh * 2 * DHD;
  const float* a2 = a1 + DHD;
  float s1 = 0.f, s2 = 0.f;
#pragma unroll
  for (int d = 0; d < DHD; ++d) { float v = w[d]; s1 += v * a1[d]; s2 += v * a2[d]; }
  s_src[idx] = s1;
  s_dst[idx] = s2;
}

// --------------------------- k3: init accumulators --------------------------
__global__ __launch_bounds__(256) void k3_init(
    float* __restrict__ m, float* __restrict__ denom, float* __restrict__ hp) {
  int idx = blockIdx.x * 256 + threadIdx.x;
  if (idx < NN * FDIM) hp[idx] = 0.f;
  if (idx < NN * NH) { m[idx] = -INFINITY; denom[idx] = 0.f; }
}

// --------------------------- k4: segment max --------------------------------
__global__ __launch_bounds__(256) void k4_segmax(
    const int* __restrict__ src, const int* __restrict__ dst,
    const float* __restrict__ s_src, const float* __restrict__ s_dst,
    float* __restrict__ m) {
  int idx = blockIdx.x * 256 + threadIdx.x;
  if (idx >= NE * NH) return;
  int e = idx >> 3, hh = idx & 7;
  int s = src[e], d = dst[e];
  float v = lrelu(s_src[s * NH + hh] + s_dst[d * NH + hh]);
  atomicMaxF(&m[d * NH + hh], v);
}

// --------------------------- k5: segment sum of exp -------------------------
__global__ __launch_bounds__(256) void k5_segsum(
    const int* __restrict__ src, const int* __restrict__ dst,
    const float* __restrict__ s_src, const float* __restrict__ s_dst,
    const float* __restrict__ m, float* __restrict__ denom) {
  int idx = blockIdx.x * 256 + threadIdx.x;
  if (idx >= NE * NH) return;
  int e = idx >> 3, hh = idx & 7;
  int s = src[e], d = dst[e];
  float v  = lrelu(s_src[s * NH + hh] + s_dst[d * NH + hh]);
  float ex = __expf(v - m[d * NH + hh]);
  atomicAdd(&denom[d * NH + hh], ex);
}

// --------------------------- k6: weighted scatter ---------------------------
// One wave per edge; lane owns 4 contiguous dims (single head per lane).
// src/dst are wave-uniform -> scalarizable; Wh gather is one b128 per lane.
__global__ __launch_bounds__(256) void k6_scatter(
    const int* __restrict__ src, const int* __restrict__ dst,
    const float* __restrict__ s_src, const float* __restrict__ s_dst,
    const float* __restrict__ m, const float* __restrict__ denom,
    const float* __restrict__ Wh, float* __restrict__ hp) {
  int tid  = threadIdx.x;
  int wave = tid >> 5, lane = tid & 31;
  int e = blockIdx.x * 8 + wave;            // 100000 blocks x 8 edges
  int s = src[e], d = dst[e];
  int hh = lane >> 2;                       // head of this lane's 4 dims
  float v    = lrelu(s_src[s * NH + hh] + s_dst[d * NH + hh]);
  float attn = __expf(v - m[d * NH + hh]) / denom[d * NH + hh];
  float4 wv  = *(const float4*)&Wh[s * FDIM + lane * 4];
  float* hpp = &hp[d * FDIM + lane * 4];
  atomicAdd(hpp + 0, attn * wv.x);
  atomicAdd(hpp + 1, attn * wv.y);
  atomicAdd(hpp + 2, attn * wv.z);
  atomicAdd(hpp + 3, attn * wv.w);
}

// --------------------------- k7: fused GRU ----------------------------------
// Each wave owns output columns [16w,16w+16) and keeps all six 16x16 gate
// tiles (i_r,i_z,i_n ; h_r,h_z,h_n) in registers; gates elementwise on
// matching D-fragments. Gate-interleaved weights -> one b96 frag per row.
__global__ __launch_bounds__(256) void k7_gru(
    const float* __restrict__ h, const float* __restrict__ hp,
    const float* __restrict__ wiP, const float* __restrict__ whP,
    const float* __restrict__ b_ih, const float* __restrict__ b_hh,
    float* __restrict__ out) {
  __shared__ __align__(16) float hA[16][132];
  __shared__ __align__(16) float pA[16][132];
  int tid = threadIdx.x;
  int mbase = blockIdx.x * 16;
  for (int i = tid; i < 512; i += 256) {
    int r = i >> 5, c4 = (i & 31) << 2;
    *(float4*)&hA[r][c4] = *(const float4*)&h [(mbase + r) * FDIM + c4];
    *(float4*)&pA[r][c4] = *(const float4*)&hp[(mbase + r) * FDIM + c4];
  }
  __syncthreads();

  int wave = tid >> 5, lane = tid & 31;
  int half = lane >> 4, l16 = lane & 15;
  int c0 = wave * 16;
  v8f ir = {}, iz = {}, inn = {}, hr = {}, hz = {}, hn = {};
  for (int k0 = 0; k0 < FDIM; k0 += 4) {
    int ks = k0 + 2 * half;
    v2f ap, ah, b;
    ap.x = pA[l16][ks];  ap.y = pA[l16][ks + 1];
    ah.x = hA[l16][ks];  ah.y = hA[l16][ks + 1];
    const float* pi0 = wiP + (ks * FDIM + c0 + l16) * 3;   // gates r,z,n
    const float* pi1 = pi0 + FDIM * 3;                     // row ks+1
    const float* ph0 = whP + (ks * FDIM + c0 + l16) * 3;
    const float* ph1 = ph0 + FDIM * 3;
    b.x = pi0[0];  b.y = pi1[0];  ir  = wmma_f32(ap, b, ir);
    b.x = pi0[1];  b.y = pi1[1];  iz  = wmma_f32(ap, b, iz);
    b.x = pi0[2];  b.y = pi1[2];  inn = wmma_f32(ap, b, inn);
    b.x = ph0[0];  b.y = ph1[0];  hr  = wmma_f32(ah, b, hr);
    b.x = ph0[1];  b.y = ph1[1];  hz  = wmma_f32(ah, b, hz);
    b.x = ph0[2];  b.y = ph1[2];  hn  = wmma_f32(ah, b, hn);
  }
  float bir = b_ih[c0 + l16], biz = b_ih[128 + c0 + l16], bin = b_ih[256 + c0 + l16];
  float bhr = b_hh[c0 + l16], bhz = b_hh[128 + c0 + l16], bhn = b_hh[256 + c0 + l16];
#pragma unroll
  for (int r = 0; r < 8; ++r) {
    int M = r + 8 * half;
    float rg = sigm(ir[r] + bir + hr[r] + bhr);
    float zg = sigm(iz[r] + biz + hz[r] + bhz);
    float ng = tanh_fast(inn[r] + bin + rg * (hn[r] + bhn));
    float hv = hA[M][c0 + l16];
    out[(mbase + M) * FDIM + c0 + l16] = (1.f - zg) * ng + zg * hv;
  }
}

// ---------------------------------------------------------------------------
extern "C" void kernel_launch(void* const* d_in, const int* in_sizes, int n_in,
                              void* d_out, int out_size, void* d_ws, size_t ws_size,
                              hipStream_t stream) {
  const float* h    = (const float*)d_in[0];
  const int*   src  = (const int*)d_in[1];
  const int*   dst  = (const int*)d_in[2];
  const float* W    = (const float*)d_in[3];
  const float* a    = (const float*)d_in[4];
  const float* w_ih = (const float*)d_in[5];
  const float* w_hh = (const float*)d_in[6];
  const float* b_ih = (const float*)d_in[7];
  const float* b_hh = (const float*)d_in[8];
  float* out = (float*)d_out;

  // workspace layout (floats); ~58 MB total
  float* ws    = (float*)d_ws;
  float* Wflat = ws;                  // 128*128      = 16384
  float* wiP   = Wflat + 16384;       // 128*384      = 49152
  float* whP   = wiP + 49152;         // 128*384      = 49152
  float* s_src = whP + 49152;         // N*8          = 400000
  float* s_dst = s_src + 400000;      // 400000
  float* m     = s_dst + 400000;      // 400000
  float* denom = m + 400000;          // 400000
  float* Wh    = denom + 400000;      // N*128        = 6400000
  float* hp    = Wh + 6400000;        // 6400000

  k0_pack   <<<192,    256, 0, stream>>>(W, w_ih, w_hh, Wflat, wiP, whP);
  k1_wh     <<<3125,   256, 0, stream>>>(h, Wflat, Wh);
  k2_scores <<<1563,   256, 0, stream>>>(Wh, a, s_src, s_dst);
  k3_init   <<<25000,  256, 0, stream>>>(m, denom, hp);
  k4_segmax <<<25000,  256, 0, stream>>>(src, dst, s_src, s_dst, m);
  k5_segsum <<<25000,  256, 0, stream>>>(src, dst, s_src, s_dst, m, denom);
  k6_scatter<<<100000, 256, 0, stream>>>(src, dst, s_src, s_dst, m, denom, Wh, hp);
  k7_gru    <<<3125,   256, 0, stream>>>(h, hp, wiP, whP, b_ih, b_hh, out);
}